// RelPositionMultiheadAttentionWeights_50379966382925
// MI455X (gfx1250) — compile-verified
//
#include <hip/hip_runtime.h>
#include <hip/hip_bf16.h>

typedef __attribute__((ext_vector_type(2))) float v2f;
typedef __attribute__((ext_vector_type(4))) float v4f;
typedef __attribute__((ext_vector_type(8))) float v8f;

namespace {
constexpr int S   = 1024;
constexpr int B   = 8;
constexpr int H   = 8;
constexpr int QH  = 32;
constexpr int PH  = 4;
constexpr int ED  = 512;
constexpr int PD  = 192;
constexpr int IPD = 544;          // (2*32+4)*8
constexpr int NPOS = 2 * S - 1;   // 2047
constexpr int ROWSTRIDE = 1028;   // padded LDS row stride (floats) -> bank-conflict free
}

// ---------------------------------------------------------------------------
// Kernel 1: xp = x @ W_in + b_in   (M=8192, K=512, N=544), fp32 WMMA 16x16x4
// One wave computes a 16(M) x 32(N) tile. 512 M-tiles * 17 N-pairs = 8704 waves.
// ---------------------------------------------------------------------------
__global__ __launch_bounds__(256) void inproj_gemm_kernel(
    const float* __restrict__ x, const float* __restrict__ W,
    const float* __restrict__ bias, float* __restrict__ xp)
{
  const int gwave = (blockIdx.x * 256 + threadIdx.x) >> 5;
  const int lane  = threadIdx.x & 31;
  const int mt = gwave / 17;
  const int np = gwave % 17;
  const int m0 = mt << 4;
  const int n0 = np << 5;
  const int mrow = lane & 15;     // A: M index
  const int hi   = lane >> 4;     // A/B: K sub-group (K = 2*hi + {0,1})
  const int ncol = lane & 15;     // B/C/D: N index

  const float* xrow = x + (size_t)(m0 + mrow) * ED + hi * 2;
  v8f c0 = {}; v8f c1 = {};
#pragma unroll 4
  for (int k0 = 0; k0 < ED; k0 += 4) {
    v2f a = *(const v2f*)(xrow + k0);                 // A[m][k0+2hi+{0,1}]
    const float* wp = W + (size_t)(k0 + hi * 2) * IPD + n0 + ncol;
    v2f b0, b1;
    b0.x = wp[0];       b0.y = wp[IPD];               // B[k][n], B[k+1][n]
    b1.x = wp[16];      b1.y = wp[IPD + 16];
    c0 = __builtin_amdgcn_wmma_f32_16x16x4_f32(false, a, false, b0, (short)0, c0, false, false);
    c1 = __builtin_amdgcn_wmma_f32_16x16x4_f32(false, a, false, b1, (short)0, c1, false, false);
  }
#pragma unroll
  for (int r = 0; r < 8; ++r) {
    const int m = m0 + r + 8 * hi;                    // C/D layout: M = r + 8*hi
    float* orow = xp + (size_t)m * IPD + n0;
    orow[ncol]      = c0[r] + bias[n0 + ncol];
    orow[16 + ncol] = c1[r] + bias[n0 + 16 + ncol];
  }
}

// ---------------------------------------------------------------------------
// Kernel 2: pe = pos_emb @ W_pos   (2047 x 192) @ (192 x 32) -> (2047, 32)
// ---------------------------------------------------------------------------
__global__ __launch_bounds__(256) void posproj_kernel(
    const float* __restrict__ pos_emb, const float* __restrict__ W_pos,
    float* __restrict__ pe)
{
  const int idx = blockIdx.x * 256 + threadIdx.x;
  if (idx >= NPOS * H * PH) return;
  const int n = idx >> 5;
  const int c = idx & 31;
  const float* pr = pos_emb + (size_t)n * PD;
  const float* wc = W_pos + c;
  float acc = 0.f;
#pragma unroll 8
  for (int j = 0; j < PD; ++j) acc = fmaf(pr[j], wc[(size_t)j * (H * PH)], acc);
  pe[idx] = acc;
}

// ---------------------------------------------------------------------------
// Kernel 3: fused  scores = Q K^T + rel_shift(pos) , mask, softmax, write out.
// Grid (hb=64, rowblk=32), block = 64 threads = 2 waves. Each wave owns 16
// query rows x 1024 keys; raw scores staged in LDS, softmax in registers.
// ---------------------------------------------------------------------------
__global__ __launch_bounds__(64) void attn_softmax_kernel(
    const float* __restrict__ xp, const float* __restrict__ pe,
    const unsigned char* __restrict__ mask, float* __restrict__ out)
{
  extern __shared__ float smem[];                     // 2 * 16 * ROWSTRIDE floats
  const int hb  = blockIdx.x;                         // h*B + b
  const int h   = hb >> 3;
  const int b   = hb & 7;
  const int wave = threadIdx.x >> 5;
  const int lane = threadIdx.x & 31;
  const int t0 = blockIdx.y * 32 + wave * 16;
  const int mrow = lane & 15;
  const int hi   = lane >> 4;
  const int ncol = lane & 15;
  float* srow = smem + wave * 16 * ROWSTRIDE;

  // Preload Q fragments for this wave's 16 rows (K = 0..31 in 8 steps of 4)
  const float* qbase = xp + ((size_t)(t0 + mrow) * B + b) * IPD + h * QH + hi * 2;
  v2f afr[8];
#pragma unroll
  for (int kk = 0; kk < 8; ++kk) afr[kk] = *(const v2f*)(qbase + kk * 4);

  // Preload p vectors for the 8 rows this lane produces (C layout rows)
  v4f pv[8];
#pragma unroll
  for (int r = 0; r < 8; ++r) {
    const int t = t0 + r + 8 * hi;
    pv[r] = *(const v4f*)(xp + ((size_t)t * B + b) * IPD + 2 * (H * QH) + h * PH);
  }

  for (int jt = 0; jt < 64; ++jt) {
    const int s0 = jt << 4;
    const int s  = s0 + ncol;
    const float* kbase = xp + ((size_t)(s0 + ncol) * B + b) * IPD + (H * QH) + h * QH + hi * 2;
    v8f c = {};
#pragma unroll
    for (int kk = 0; kk < 8; ++kk) {
      v2f bfr = *(const v2f*)(kbase + kk * 4);        // B[k][n] = K[s0+n][k]
      c = __builtin_amdgcn_wmma_f32_16x16x4_f32(false, afr[kk], false, bfr, (short)0, c, false, false);
    }
    const bool msk = mask[b * S + s] != 0;
#pragma unroll
    for (int r = 0; r < 8; ++r) {
      const int m = r + 8 * hi;
      const int t = t0 + m;
      float val = -1000.0f;
      if (!msk) {
        // relative shift: pos index = (S-1) - t + s, always in [0, 2S-2]
        const v4f pev = *(const v4f*)(pe + (size_t)((S - 1) - t + s) * (H * PH) + h * PH);
        val = c[r] + pv[r].x * pev.x + pv[r].y * pev.y + pv[r].z * pev.z + pv[r].w * pev.w;
      }
      srow[m * ROWSTRIDE + s] = val;
    }
  }
  __syncthreads();

  // Row softmax: 32 lanes cooperate per row; 32 values/lane stay in registers.
  for (int r = 0; r < 16; ++r) {
    const float* row = srow + r * ROWSTRIDE;
    float v[32];
    float mx = -3.0e38f;
#pragma unroll
    for (int i = 0; i < 32; ++i) { v[i] = row[i * 32 + lane]; mx = fmaxf(mx, v[i]); }
#pragma unroll
    for (int off = 16; off > 0; off >>= 1) mx = fmaxf(mx, __shfl_xor(mx, off, 32));
    float sum = 0.f;
#pragma unroll
    for (int i = 0; i < 32; ++i) { v[i] = __expf(v[i] - mx); sum += v[i]; }
#pragma unroll
    for (int off = 16; off > 0; off >>= 1) sum += __shfl_xor(sum, off, 32);
    const float inv = 1.0f / sum;
    float* orow = out + (((size_t)hb) * S + (t0 + r)) * S;
#pragma unroll
    for (int i = 0; i < 32; ++i) orow[i * 32 + lane] = v[i] * inv;
  }
}

// ---------------------------------------------------------------------------
extern "C" void kernel_launch(void* const* d_in, const int* in_sizes, int n_in,
                              void* d_out, int out_size, void* d_ws, size_t ws_size,
                              hipStream_t stream) {
  const float* x      = (const float*)d_in[0];   // (S, B, 512)
  const float* posemb = (const float*)d_in[1];   // (1, 2S-1, 192)
  const float* W_in   = (const float*)d_in[2];   // (512, 544)
  const float* b_in   = (const float*)d_in[3];   // (544,)
  const float* W_pos  = (const float*)d_in[4];   // (192, 32)
  const unsigned char* mask = (const unsigned char*)d_in[5]; // (B, S) bool
  float* out = (float*)d_out;                    // (H, B, S, S)

  float* xp = (float*)d_ws;                      // (S*B, 544)  ~17.8 MB
  float* pe = xp + (size_t)(S * B) * IPD;        // (2047, 32)  ~256 KB

  // xp = x @ W_in + b_in
  inproj_gemm_kernel<<<(512 * 17 * 32) / 256, 256, 0, stream>>>(x, W_in, b_in, xp);
  // pe = pos_emb @ W_pos
  posproj_kernel<<<(NPOS * H * PH + 255) / 256, 256, 0, stream>>>(posemb, W_pos, pe);
  // fused scores + rel-shift + mask + softmax
  dim3 g3(H * B, S / 32);
  attn_softmax_kernel<<<g3, 64, 2 * 16 * ROWSTRIDE * sizeof(float), stream>>>(xp, pe, mask, out);
}